// AssociativeAttention_14104672600709
// MI455X (gfx1250) — compile-verified
//
#include <hip/hip_runtime.h>
#include <math.h>

// ---------------- problem constants (from reference setup) ----------------
constexpr int Bc   = 2;
constexpr int Hc   = 8;
constexpr int Lc   = 2048;
constexpr int Dc   = 256;
constexpr int hc   = 32;       // head dim
constexpr int CH   = 32;       // chunk length for the scan
constexpr int NCH  = Lc / CH;  // 64 chunks
constexpr float EPSc = 1e-5f;

typedef float v2f __attribute__((ext_vector_type(2)));
typedef float v8f __attribute__((ext_vector_type(8)));

// wave32 full reduction
__device__ inline float wsum(float x) {
  #pragma unroll
  for (int m = 16; m > 0; m >>= 1) x += __shfl_xor(x, m, 32);
  return x;
}

// =====================================================================
// Kernel 1: fused Q/K/V projection.  X[4096,256] * W[n,256]^T + b.
// 4 waves per block, 32x32 macro-tile (each wave one 16x16 quadrant),
// fp32 WMMA 16x16x4 over K=256.  Output in [b, head, l, d] layout.
// =====================================================================
__global__ __launch_bounds__(128) void k_gemm_qkv(
    const float* __restrict__ X,
    const float* __restrict__ wq, const float* __restrict__ bq,
    const float* __restrict__ wk, const float* __restrict__ bk,
    const float* __restrict__ wv, const float* __restrict__ bv,
    float* __restrict__ qb, float* __restrict__ kb, float* __restrict__ vb)
{
  const int tid  = threadIdx.x;
  const int wave = tid >> 5;
  const int lane = tid & 31;
  const int mt = blockIdx.x & 127;   // 128 macro M tiles (M = 4096)
  const int nt = blockIdx.x >> 7;    // 24 macro N tiles (q:0-7, k:8-15, v:16-23)
  const int which = nt >> 3;
  const int o0 = (nt & 7) << 5;      // base output col within selected matrix
  const float* W    = (which == 0) ? wq : (which == 1) ? wk : wv;
  const float* bias = (which == 0) ? bq : (which == 1) ? bk : bv;
  float*       O    = (which == 0) ? qb : (which == 1) ? kb : vb;

  __shared__ float At[32][260];   // pad 260: kills 16-way bank conflicts
  __shared__ float Bt[32][260];
  const int m0 = mt << 5;
  for (int i = tid; i < 32 * 256; i += 128) {
    int r = i >> 8, c = i & 255;
    At[r][c] = X[(m0 + r) * 256 + c];
    Bt[r][c] = W[(o0 + r) * 256 + c];
  }
  __syncthreads();

  v8f acc;
  #pragma unroll
  for (int r = 0; r < 8; ++r) acc[r] = 0.0f;

  // quadrant assignment: wave bit0 -> N half, bit1 -> M half
  const int mrow = (lane & 15) + ((wave >> 1) << 4);
  const int ncol = (lane & 15) + ((wave & 1) << 4);
  const int kk   = (lane >> 4) << 1;
  #pragma unroll 4
  for (int k0 = 0; k0 < 256; k0 += 4) {
    v2f a, b;
    a[0] = At[mrow][k0 + kk];
    a[1] = At[mrow][k0 + kk + 1];
    b[0] = Bt[ncol][k0 + kk];       // B[k][n] = W[n][k]
    b[1] = Bt[ncol][k0 + kk + 1];
    acc = __builtin_amdgcn_wmma_f32_16x16x4_f32(
        false, a, false, b, (short)0, acc, false, false);
  }

  // D layout: VGPR r -> (M=r, N=lane) lanes 0-15 ; (M=r+8, N=lane-16)
  const int nn = (lane & 15) + ((wave & 1) << 4);
  const int mb = ((lane >> 4) << 3) + ((wave >> 1) << 4);
  #pragma unroll
  for (int r = 0; r < 8; ++r) {
    int m = m0 + mb + r;
    int o = o0 + nn;
    float val = acc[r] + bias[o];
    int bI = m >> 11, l = m & 2047;
    int head = o >> 5, d = o & 31;
    O[(((bI * Hc + head) * Lc + l) << 5) + d] = val;
  }
}

// =====================================================================
// Kernel 2 (pass 1): per-position prep + per-chunk commutative totals.
// One block = one (b, head, chunk); 8 waves, each wave owns CH/8 positions.
// Writes: normalized k,v (in place), sim, gate, chunk totals (gZ, g, m, s).
// =====================================================================
__global__ __launch_bounds__(256) void k_pass1(
    const float* __restrict__ qb, float* __restrict__ kb, float* __restrict__ vb,
    const float* __restrict__ wg_w, const float* __restrict__ wg_b,
    const float* __restrict__ kvs, const float* __restrict__ qks,
    float* __restrict__ simb, float* __restrict__ gateb,
    float* __restrict__ cgZ, float* __restrict__ cg,
    float* __restrict__ cm, float* __restrict__ cs)
{
  const int tid  = threadIdx.x;
  const int wave = tid >> 5;
  const int lane = tid & 31;
  const int c    = blockIdx.x & (NCH - 1);
  const int bh   = blockIdx.x >> 6;        // b*Hc + head
  const int head = bh & (Hc - 1);

  __shared__ float weff[32 * 33];          // wg * kv_scale, padded rows
  __shared__ float kvl[32 * 32];           // kv_scale for this head
  __shared__ float red[8 * 1024];          // per-wave gZ partials
  __shared__ float sg[8], sm[8], ss[8];

  for (int i = tid; i < 1024; i += 256) {
    float kvv = kvs[head * 1024 + i];
    kvl[i] = kvv;
    weff[(i >> 5) * 33 + (i & 31)] = wg_w[i] * kvv;
  }
  __syncthreads();

  const float qscale = qks[head];
  const float wgb = wg_b[0];

  float zc[32];                            // lane = column e
  #pragma unroll
  for (int d = 0; d < 32; ++d) zc[d] = 0.0f;
  float gsum = 0.0f, mrun = -1e30f, srun = 0.0f;

  for (int p = wave; p < CH; p += 8) {
    const int l   = c * CH + p;
    const int idx = ((bh * Lc + l) << 5) + lane;
    float qv = qb[idx], kvr = kb[idx], vvr = vb[idx];

    float sim = wsum(qv * kvr) * qscale;                 // raw k (pre-norm)
    float kn = kvr / fmaxf(sqrtf(wsum(kvr * kvr)), 1e-12f);
    float vn = vvr / fmaxf(sqrtf(wsum(vvr * vvr)), 1e-12f);
    kb[idx] = kn;
    vb[idx] = vn;

    // gate logit = sum_{d,e} vn_d * kn_e * (wg*kvs)[d,e]; lane plays role of d
    float t = 0.0f;
    #pragma unroll
    for (int e = 0; e < 32; ++e)
      t = fmaf(weff[lane * 33 + e], __shfl(kn, e, 32), t);
    float logit = wsum(vn * t) + wgb;
    float g = fmaxf(logit, 0.0f);
    g = g * g + EPSc;

    if (lane == 0) { simb[bh * Lc + l] = sim; gateb[bh * Lc + l] = g; }

    // accumulate chunk totals (commutative)
    float gk = g * kn;
    #pragma unroll
    for (int d = 0; d < 32; ++d)
      zc[d] = fmaf(gk * kvl[d * 32 + lane], __shfl(vn, d, 32), zc[d]);
    float mn = fmaxf(mrun, sim);
    srun = srun * __expf(mrun - mn) + __expf(sim - mn);
    mrun = mn;
    gsum += g;
  }

  // cross-wave reduction of totals
  #pragma unroll
  for (int d = 0; d < 32; ++d) red[wave * 1024 + d * 32 + lane] = zc[d];
  if (lane == 0) { sg[wave] = gsum; sm[wave] = mrun; ss[wave] = srun; }
  __syncthreads();

  for (int i = tid; i < 1024; i += 256) {
    float a = 0.0f;
    #pragma unroll
    for (int w = 0; w < 8; ++w) a += red[w * 1024 + i];
    cgZ[(size_t)blockIdx.x * 1024 + i] = a;
  }
  if (tid == 0) {
    float M = -1e30f, S = 0.0f, G = 0.0f;
    #pragma unroll
    for (int w = 0; w < 8; ++w) {
      float Mn = fmaxf(M, sm[w]);
      S = S * __expf(M - Mn) + ss[w] * __expf(sm[w] - Mn);
      M = Mn;
      G += sg[w];
    }
    cm[blockIdx.x] = M; cs[blockIdx.x] = S; cg[blockIdx.x] = G;
  }
}

// =====================================================================
// Kernel 3: exclusive scan of chunk totals per (b,head). 16 blocks.
// =====================================================================
__global__ __launch_bounds__(256) void k_chunkscan(
    const float* __restrict__ cgZ, const float* __restrict__ cg,
    const float* __restrict__ cm, const float* __restrict__ cs,
    float* __restrict__ gZp, float* __restrict__ gp,
    float* __restrict__ mp, float* __restrict__ sp)
{
  const int bh = blockIdx.x;
  const int tid = threadIdx.x;
  for (int i = tid; i < 1024; i += 256) {
    float acc = 0.0f;
    for (int c = 0; c < NCH; ++c) {
      size_t o = (size_t)(bh * NCH + c) * 1024 + i;
      gZp[o] = acc;
      acc += cgZ[o];
    }
  }
  if (tid == 0) {
    float M = -1e30f, S = 0.0f, G = 0.0f;
    for (int c = 0; c < NCH; ++c) {
      int o = bh * NCH + c;
      mp[o] = M; sp[o] = S; gp[o] = G;
      float Mn = fmaxf(M, cm[o]);
      S = S * __expf(M - Mn) + cs[o] * __expf(cm[o] - Mn);
      M = Mn;
      G += cg[o];
    }
  }
}

// =====================================================================
// Kernel 4 (pass 2): sequential replay of each chunk with the 32x32
// running state one-column-per-lane.  Whole chunk is staged to LDS first
// (coalesced), so the serial loop runs out of LDS broadcasts (no shfl,
// no global latency in the dependence chain).  1024 independent waves.
// =====================================================================
__global__ __launch_bounds__(32) void k_pass2(
    const float* __restrict__ qb, const float* __restrict__ kb,
    const float* __restrict__ vb, const float* __restrict__ kvs,
    const float* __restrict__ simb, const float* __restrict__ gateb,
    const float* __restrict__ gZp, const float* __restrict__ gp,
    const float* __restrict__ mp, const float* __restrict__ sp,
    float* __restrict__ ctxt)
{
  const int lane = threadIdx.x;
  const int c    = blockIdx.x & (NCH - 1);
  const int bh   = blockIdx.x >> 6;
  const int head = bh & (Hc - 1);
  const int b    = bh >> 3;

  __shared__ float qs[CH * 32], ks[CH * 32], vs[CH * 32];
  __shared__ float sims[CH], gats[CH];

  // stage the chunk: positions are contiguous in [b,h,l,d] layout
  const size_t base = ((size_t)(bh * Lc + c * CH)) << 5;
  for (int i = lane; i < CH * 32; i += 32) {
    qs[i] = qb[base + i];
    ks[i] = kb[base + i];
    vs[i] = vb[base + i];
  }
  if (lane < CH) {
    sims[lane] = simb[bh * Lc + c * CH + lane];
    gats[lane] = gateb[bh * Lc + c * CH + lane];
  }
  __syncthreads();

  float zc[32], kvc[32];
  #pragma unroll
  for (int d = 0; d < 32; ++d) {
    zc[d]  = gZp[(size_t)(bh * NCH + c) * 1024 + d * 32 + lane];
    kvc[d] = kvs[head * 1024 + d * 32 + lane];
  }
  float m = mp[bh * NCH + c];
  float s = sp[bh * NCH + c];
  float g = gp[bh * NCH + c];

  for (int p = 0; p < CH; ++p) {
    float kn   = ks[p * 32 + lane];
    float sim  = sims[p];
    float gate = gats[p];

    float mn = fmaxf(m, sim);                         // inclusive scans
    s = s * __expf(m - mn) + __expf(sim - mn);
    m = mn;
    g += gate;

    float gk = gate * kn;
    float acc = 0.0f;
    #pragma unroll
    for (int d = 0; d < 32; ++d) {
      float vd = vs[p * 32 + d];                      // LDS broadcast
      float qd = qs[p * 32 + d];                      // LDS broadcast
      zc[d] = fmaf(gk * kvc[d], vd, zc[d]);           // Z_scan update
      acc = fmaf(qd, zc[d], acc);                     // q . Z_scan[:,e]
    }

    float sw = __expf(sim - m) / (s + EPSc);
    float silu = sw / (1.0f + __expf(-sw));
    float factor = (1.0f + silu) / (g + EPSc);
    const int l = c * CH + p;
    ctxt[((size_t)(b * Lc + l) << 8) + (head << 5) + lane] = acc * factor;
  }
}

// =====================================================================
// Kernel 5: output projection ctxt[4096,256] * wo^T + bo -> d_out
// 4 waves / 32x32 macro-tile, same WMMA feed as kernel 1.
// =====================================================================
__global__ __launch_bounds__(128) void k_gemm_out(
    const float* __restrict__ X, const float* __restrict__ W,
    const float* __restrict__ bias, float* __restrict__ out)
{
  const int tid  = threadIdx.x;
  const int wave = tid >> 5;
  const int lane = tid & 31;
  const int mt = blockIdx.x & 127;
  const int nt = blockIdx.x >> 7;     // 8 macro N tiles
  const int o0 = nt << 5;

  __shared__ float At[32][260];
  __shared__ float Bt[32][260];
  const int m0 = mt << 5;
  for (int i = tid; i < 32 * 256; i += 128) {
    int r = i >> 8, cidx = i & 255;
    At[r][cidx] = X[(m0 + r) * 256 + cidx];
    Bt[r][cidx] = W[(o0 + r) * 256 + cidx];
  }
  __syncthreads();

  v8f acc;
  #pragma unroll
  for (int r = 0; r < 8; ++r) acc[r] = 0.0f;

  const int mrow = (lane & 15) + ((wave >> 1) << 4);
  const int ncol = (lane & 15) + ((wave & 1) << 4);
  const int kk   = (lane >> 4) << 1;
  #pragma unroll 4
  for (int k0 = 0; k0 < 256; k0 += 4) {
    v2f a, bfr;
    a[0]   = At[mrow][k0 + kk];
    a[1]   = At[mrow][k0 + kk + 1];
    bfr[0] = Bt[ncol][k0 + kk];
    bfr[1] = Bt[ncol][k0 + kk + 1];
    acc = __builtin_amdgcn_wmma_f32_16x16x4_f32(
        false, a, false, bfr, (short)0, acc, false, false);
  }

  const int nn = (lane & 15) + ((wave & 1) << 4);
  const int mb = ((lane >> 4) << 3) + ((wave >> 1) << 4);
  #pragma unroll
  for (int r = 0; r < 8; ++r) {
    int m = m0 + mb + r;
    int o = o0 + nn;
    out[(size_t)m * 256 + o] = acc[r] + bias[o];
  }
}

// =====================================================================
extern "C" void kernel_launch(void* const* d_in, const int* in_sizes, int n_in,
                              void* d_out, int out_size, void* d_ws, size_t ws_size,
                              hipStream_t stream) {
  const float* x    = (const float*)d_in[0];
  const float* wq_w = (const float*)d_in[1];
  const float* wq_b = (const float*)d_in[2];
  const float* wk_w = (const float*)d_in[3];
  const float* wk_b = (const float*)d_in[4];
  const float* wv_w = (const float*)d_in[5];
  const float* wv_b = (const float*)d_in[6];
  const float* wo_w = (const float*)d_in[7];
  const float* wo_b = (const float*)d_in[8];
  const float* wg_w = (const float*)d_in[9];
  const float* wg_b = (const float*)d_in[10];
  const float* kvs  = (const float*)d_in[11];
  const float* qks  = (const float*)d_in[12];
  float* out = (float*)d_out;

  // workspace layout (floats)
  float* ws = (float*)d_ws;
  const size_t NQ = (size_t)Bc * Hc * Lc * hc;   // 1,048,576
  float* qbuf  = ws;                // [B,H,L,h]
  float* kbuf  = qbuf + NQ;         // raw k, overwritten with normalized
  float* vbuf  = kbuf + NQ;
  float* ctxt  = vbuf + NQ;         // [B,L,D]
  float* simb  = ctxt + NQ;         // [B,H,L]
  float* gateb = simb + (size_t)Bc * Hc * Lc;
  float* cgZ   = gateb + (size_t)Bc * Hc * Lc;              // [BH*NCH,1024]
  float* gZp   = cgZ + (size_t)Bc * Hc * NCH * 1024;        // exclusive prefixes
  float* cm    = gZp + (size_t)Bc * Hc * NCH * 1024;
  float* cs    = cm + Bc * Hc * NCH;
  float* cg    = cs + Bc * Hc * NCH;
  float* mpre  = cg + Bc * Hc * NCH;
  float* spre  = mpre + Bc * Hc * NCH;
  float* gpre  = spre + Bc * Hc * NCH;

  // 1) Q/K/V projections: 128 M-macrotiles x 24 N-macrotiles, 4 waves each
  k_gemm_qkv<<<128 * 24, 128, 0, stream>>>(x, wq_w, wq_b, wk_w, wk_b, wv_w, wv_b,
                                           qbuf, kbuf, vbuf);
  // 2) per-position prep + chunk totals: (B*H*NCH) blocks of 256
  k_pass1<<<Bc * Hc * NCH, 256, 0, stream>>>(qbuf, kbuf, vbuf, wg_w, wg_b, kvs,
                                             qks, simb, gateb, cgZ, cg, cm, cs);
  // 3) exclusive scan of chunk totals per (b,head)
  k_chunkscan<<<Bc * Hc, 256, 0, stream>>>(cgZ, cg, cm, cs, gZp, gpre, mpre, spre);
  // 4) sequential chunk replay -> ctxt (1024 independent waves)
  k_pass2<<<Bc * Hc * NCH, 32, 0, stream>>>(qbuf, kbuf, vbuf, kvs, simb, gateb,
                                            gZp, gpre, mpre, spre, ctxt);
  // 5) output projection
  k_gemm_out<<<128 * 8, 128, 0, stream>>>(ctxt, wo_w, wo_b, out);
}